// Convolution_v0_13099650253152
// MI455X (gfx1250) — compile-verified
//
#include <hip/hip_runtime.h>
#include <math.h>

// ---------------------------------------------------------------------------
// Fused e3nn-style equivariant convolution for gfx1250 (MI455X).
//   node_pos [8192,16] f32, edge_type [65536,16] f32,
//   fc_w0 [16,64] f32, fc_w1 [64,2560] f32, edge_src/dst [65536] i32
//   out [8192,16] f32
// FC2 (hidden[E,64] @ fc_w1[64,2560]) is 99.5% of the FLOPs -> WMMA f32.
// Fusion avoids the 671MB per-edge-weight intermediate entirely.
// FC2 K-reduction split into 2 independent WMMA chains for XDL pipelining.
// ---------------------------------------------------------------------------

typedef float v2f __attribute__((ext_vector_type(2)));
typedef float v8f __attribute__((ext_vector_type(8)));

#define INV_SQRT2  0.70710678f
#define INV_SQRT3  0.57735027f
#define INV_SQRT6  0.40824829f
#define INV_SQRT10 0.31622777f
#define SILU_C     1.67668f      // e3nn normalize2mom(silu)
#define SC0        0.02209709f   // path_w(1/16)   * 1/sqrt(8)
#define SC1        0.03125f      // path_w(1/8√2) * 1/sqrt(8)

__global__ void k_zero(float* p, int n) {
  int i = blockIdx.x * blockDim.x + threadIdx.x;
  if (i < n) p[i] = 0.f;
}

// Repack fc_w1 [64,2560] into WMMA-B fragment order in d_ws:
//   ws[((t*16+s)*32 + lane)*2 + c] = fc_w1[4s + (lane<16?0:2) + c][t*16 + lane%16]
// so the FC2 inner loop does one coalesced b64 load per lane per K-step.
__global__ void k_repack_w1(const float* __restrict__ w1, float* __restrict__ o) {
  int tid = blockIdx.x * blockDim.x + threadIdx.x;
  if (tid >= 160 * 16 * 32) return;
  int l  = tid & 31;
  int s  = (tid >> 5) & 15;
  int t  = tid >> 9;
  int li = l & 15;
  int kd = (l < 16) ? 0 : 2;
  int row = 4 * s + kd;
  int col = t * 16 + li;
  o[2 * tid + 0] = w1[row * 2560 + col];
  o[2 * tid + 1] = w1[(row + 1) * 2560 + col];
}

__global__ __launch_bounds__(32)
void k_edge_conv(const float* __restrict__ node_pos,
                 const float* __restrict__ edge_type,
                 const float* __restrict__ fc_w0,
                 const float* __restrict__ w1p,       // repacked fc_w1 in d_ws
                 const int*   __restrict__ edge_src,
                 const int*   __restrict__ edge_dst,
                 float* __restrict__ out)
{
  __shared__ float hid[16 * 66];    // hidden [16 edges][64], padded stride 66
  __shared__ float xsr[16 * 20];    // x_src rows  [4 scalars | 4x3 vectors]
  __shared__ float xds[16 * 20];    // x_dst rows
  __shared__ float mid[16 * 260];   // TP1 output [16 edges][256], padded
  __shared__ int   dsti[16];

  const int lane = threadIdx.x;          // 0..31 (wave32)
  const int li   = lane & 15;
  const int kd   = (lane < 16) ? 0 : 2;  // K-pair selector for A/B fragments
  const int eoff = (lane < 16) ? 0 : 8;  // D-row offset (M = r + eoff)
  const int e0   = blockIdx.x * 16;

  // -------- gather node rows (lanes 0-15: src, lanes 16-31: dst) ----------
  if (lane < 16) {
    int s = edge_src[e0 + li];
    dsti[li] = edge_dst[e0 + li];
    const float4* q = (const float4*)(node_pos + (size_t)s * 16);
    float4* d = (float4*)(xsr + li * 20);
    d[0] = q[0]; d[1] = q[1]; d[2] = q[2]; d[3] = q[3];
  } else {
    int dd = edge_dst[e0 + li];
    const float4* q = (const float4*)(node_pos + (size_t)dd * 16);
    float4* d = (float4*)(xds + li * 20);
    d[0] = q[0]; d[1] = q[1]; d[2] = q[2]; d[3] = q[3];
  }

  // -------- FC1: hidden = silu(edge_type @ fc_w0 / 4)*C / 8  (WMMA) -------
  {
    v2f a[4];
    const float* et = edge_type + (size_t)(e0 + li) * 16 + kd;
#pragma unroll
    for (int s = 0; s < 4; ++s) a[s] = *(const v2f*)(et + 4 * s);
#pragma unroll
    for (int tt = 0; tt < 4; ++tt) {
      v8f c = {};
#pragma unroll
      for (int s = 0; s < 4; ++s) {
        int row = 4 * s + kd;
        v2f b;
        b.x = fc_w0[row * 64 + tt * 16 + li];
        b.y = fc_w0[(row + 1) * 64 + tt * 16 + li];
        c = __builtin_amdgcn_wmma_f32_16x16x4_f32(false, a[s], false, b,
                                                  (short)0, c, false, false);
      }
#pragma unroll
      for (int r = 0; r < 8; ++r) {
        float x = c[r] * 0.25f;
        float h = (x / (1.f + __expf(-x))) * (SILU_C * 0.125f);
        hid[(r + eoff) * 66 + tt * 16 + li] = h;
      }
    }
  }
  __syncthreads();

  // -------- TP1: mid[256] per edge (2 lanes/edge, u split by wave half) ---
  {
    const int e = li;
    const float* X = xsr + e * 20;
    const float* Y = xds + e * 20;
    float y0[4], y1[4][3];
#pragma unroll
    for (int v = 0; v < 4; ++v) {
      y0[v] = Y[v];
      y1[v][0] = Y[4 + v * 3 + 0];
      y1[v][1] = Y[4 + v * 3 + 1];
      y1[v][2] = Y[4 + v * 3 + 2];
    }
    float* M = mid + e * 260;
    const int ub = (lane < 16) ? 0 : 2;
#pragma unroll
    for (int uu = 0; uu < 2; ++uu) {
      int u = ub + uu;
      float x0 = X[u];
      float a0 = X[4 + u * 3 + 0], a1 = X[4 + u * 3 + 1], a2 = X[4 + u * 3 + 2];
#pragma unroll
      for (int v = 0; v < 4; ++v) {
        int uv = u * 4 + v;
        float b0 = y1[v][0], b1 = y1[v][1], b2 = y1[v][2];
        M[uv] = x0 * y0[v];                               // 0e x 0e -> 0e
        M[16 + uv * 3 + 0] = x0 * b0;                      // 0e x 1o -> 1o
        M[16 + uv * 3 + 1] = x0 * b1;
        M[16 + uv * 3 + 2] = x0 * b2;
        M[64 + uv * 3 + 0] = a0 * y0[v];                   // 1o x 0e -> 1o
        M[64 + uv * 3 + 1] = a1 * y0[v];
        M[64 + uv * 3 + 2] = a2 * y0[v];
        M[112 + uv] = (a0 * b0 + a1 * b1 + a2 * b2) * INV_SQRT3;  // ->0e
        M[128 + uv * 3 + 0] = (a1 * b2 - a2 * b1) * INV_SQRT2;    // ->1e
        M[128 + uv * 3 + 1] = (a2 * b0 - a0 * b2) * INV_SQRT2;
        M[128 + uv * 3 + 2] = (a0 * b1 - a1 * b0) * INV_SQRT2;
        M[176 + uv * 5 + 0] = (a2 * b0 + a0 * b2) * INV_SQRT2;    // ->2e
        M[176 + uv * 5 + 1] = (a0 * b1 + a1 * b0) * INV_SQRT2;
        M[176 + uv * 5 + 2] = (2.f * a1 * b1 - a0 * b0 - a2 * b2) * INV_SQRT6;
        M[176 + uv * 5 + 3] = (a1 * b2 + a2 * b1) * INV_SQRT2;
        M[176 + uv * 5 + 4] = (a2 * b2 - a0 * b0) * INV_SQRT2;
      }
    }
  }
  __syncthreads();

  // -------- FC2 (WMMA) fused with TP2 consume -----------------------------
  v2f A[16];                         // hidden A-fragments, reused 160x
#pragma unroll
  for (int s = 0; s < 16; ++s) A[s] = *(const v2f*)&hid[li * 66 + 4 * s + kd];

  float acc0[8], acc1[8][3];
#pragma unroll
  for (int r = 0; r < 8; ++r) {
    acc0[r] = 0.f; acc1[r][0] = 0.f; acc1[r][1] = 0.f; acc1[r][2] = 0.f;
  }

#pragma unroll
  for (int p = 0; p < 10; ++p) {            // 10 uvw paths, 16 N-tiles each
    for (int tt = 0; tt < 16; ++tt) {
      int t = p * 16 + tt;
      const float* bq = w1p + (size_t)t * 1024 + 2 * lane;
      // Two independent K-chains (8 WMMAs each) so the XDL pipe can overlap
      // dependent accumulations within this wave; summed at the end.
      v8f ca = {}, cb = {};
#pragma unroll
      for (int s = 0; s < 8; ++s) {         // K = 64 in steps of 4, split 2x
        v2f b0 = *(const v2f*)(bq + (2 * s) * 64);
        v2f b1 = *(const v2f*)(bq + (2 * s + 1) * 64);
        ca = __builtin_amdgcn_wmma_f32_16x16x4_f32(false, A[2 * s], false, b0,
                                                   (short)0, ca, false, false);
        cb = __builtin_amdgcn_wmma_f32_16x16x4_f32(false, A[2 * s + 1], false, b1,
                                                   (short)0, cb, false, false);
      }
      v8f c = ca + cb;
      // lane's weight column: n = t*16 + li -> (path p, u, v, w = li&3)
      int rem = tt * 16 + li;
      int u = rem >> 6;
      int v = (rem >> 2) & 15;
#pragma unroll
      for (int r = 0; r < 8; ++r) {
        int e = r + eoff;
        float W = c[r];
        const float* X = xsr + e * 20;
        const float* M = mid + e * 260;
        if (p == 0) {                                   // 0e x mid0 -> 0e
          acc0[r] += W * X[u] * M[v];
        } else if (p == 1) {                            // 0e x mid1 -> 1o
          float f = W * X[u] * INV_SQRT3;
          acc1[r][0] += f * M[16 + v * 3 + 0];
          acc1[r][1] += f * M[16 + v * 3 + 1];
          acc1[r][2] += f * M[16 + v * 3 + 2];
        } else if (p == 2) {                            // 0e x mid2 -> 1o
          float f = W * X[u] * INV_SQRT3;
          acc1[r][0] += f * M[64 + v * 3 + 0];
          acc1[r][1] += f * M[64 + v * 3 + 1];
          acc1[r][2] += f * M[64 + v * 3 + 2];
        } else if (p == 3) {                            // 0e x mid3 -> 0e
          acc0[r] += W * X[u] * M[112 + v];
        } else if (p == 4) {                            // 1o x mid0 -> 1o
          float f = W * M[v] * INV_SQRT3;
          acc1[r][0] += f * X[4 + u * 3 + 0];
          acc1[r][1] += f * X[4 + u * 3 + 1];
          acc1[r][2] += f * X[4 + u * 3 + 2];
        } else if (p == 5) {                            // 1o . mid1 -> 0e
          float d = X[4 + u * 3 + 0] * M[16 + v * 3 + 0]
                  + X[4 + u * 3 + 1] * M[16 + v * 3 + 1]
                  + X[4 + u * 3 + 2] * M[16 + v * 3 + 2];
          acc0[r] += W * d * INV_SQRT3;
        } else if (p == 6) {                            // 1o . mid2 -> 0e
          float d = X[4 + u * 3 + 0] * M[64 + v * 3 + 0]
                  + X[4 + u * 3 + 1] * M[64 + v * 3 + 1]
                  + X[4 + u * 3 + 2] * M[64 + v * 3 + 2];
          acc0[r] += W * d * INV_SQRT3;
        } else if (p == 7) {                            // 1o x mid3 -> 1o
          float f = W * M[112 + v] * INV_SQRT3;
          acc1[r][0] += f * X[4 + u * 3 + 0];
          acc1[r][1] += f * X[4 + u * 3 + 1];
          acc1[r][2] += f * X[4 + u * 3 + 2];
        } else if (p == 8) {                            // 1o x mid4 (cross)
          float a0 = X[4 + u * 3 + 0], a1 = X[4 + u * 3 + 1], a2 = X[4 + u * 3 + 2];
          float m0 = M[128 + v * 3 + 0], m1 = M[128 + v * 3 + 1], m2 = M[128 + v * 3 + 2];
          float f = W * INV_SQRT6;
          acc1[r][0] += f * (a1 * m2 - a2 * m1);
          acc1[r][1] += f * (a2 * m0 - a0 * m2);
          acc1[r][2] += f * (a0 * m1 - a1 * m0);
        } else {                                        // 1o x mid5 (1x2->1)
          float a0 = X[4 + u * 3 + 0], a1 = X[4 + u * 3 + 1], a2 = X[4 + u * 3 + 2];
          const float* m5 = M + 176 + v * 5;
          float m0 = m5[0], m1 = m5[1], m2 = m5[2], m3 = m5[3], m4 = m5[4];
          float f = W * INV_SQRT10;
          float s2 = m2 * INV_SQRT3;
          acc1[r][0] += f * ((-s2 - m4) * a0 + m1 * a1 + m0 * a2);
          acc1[r][1] += f * (m1 * a0 + 2.f * s2 * a1 + m3 * a2);
          acc1[r][2] += f * (m0 * a0 + m3 * a1 + (-s2 + m4) * a2);
        }
      }
    }
  }

  // -------- cross-lane reduce (lanes w, w+4, w+8, w+12) + scatter-add -----
#pragma unroll
  for (int r = 0; r < 8; ++r) {
    float s0 = acc0[r];
    float t0 = acc1[r][0], t1 = acc1[r][1], t2 = acc1[r][2];
    s0 += __shfl_xor(s0, 4); s0 += __shfl_xor(s0, 8);
    t0 += __shfl_xor(t0, 4); t0 += __shfl_xor(t0, 8);
    t1 += __shfl_xor(t1, 4); t1 += __shfl_xor(t1, 8);
    t2 += __shfl_xor(t2, 4); t2 += __shfl_xor(t2, 8);
    if (li < 4) {                               // li == w
      int e = r + eoff;
      float* o = out + (size_t)dsti[e] * 16;
      atomicAdd(o + li,             s0 * SC0);
      atomicAdd(o + 4 + 3 * li + 0, t0 * SC1);
      atomicAdd(o + 4 + 3 * li + 1, t1 * SC1);
      atomicAdd(o + 4 + 3 * li + 2, t2 * SC1);
    }
  }
}

extern "C" void kernel_launch(void* const* d_in, const int* in_sizes, int n_in,
                              void* d_out, int out_size, void* d_ws, size_t ws_size,
                              hipStream_t stream) {
  const float* node_pos  = (const float*)d_in[0];
  const float* edge_type = (const float*)d_in[1];
  const float* fc_w0     = (const float*)d_in[2];
  const float* fc_w1     = (const float*)d_in[3];
  const int*   esrc      = (const int*)d_in[4];
  const int*   edst      = (const int*)d_in[5];
  float* out = (float*)d_out;
  float* w1p = (float*)d_ws;                    // 640 KB repacked fc_w1
  int E = in_sizes[4];

  k_zero<<<(out_size + 255) / 256, 256, 0, stream>>>(out, out_size);
  k_repack_w1<<<(160 * 16 * 32 + 255) / 256, 256, 0, stream>>>(fc_w1, w1p);
  k_edge_conv<<<E / 16, 32, 0, stream>>>(node_pos, edge_type, fc_w0, w1p,
                                         esrc, edst, out);
}